// SoftmaxDecoder_32487132627149
// MI455X (gfx1250) — compile-verified
//
#include <hip/hip_runtime.h>
#include <hip/hip_bf16.h>
#include <math.h>

typedef __attribute__((ext_vector_type(2))) float v2f;
typedef __attribute__((ext_vector_type(8))) float v8f;

#define DIM 64
#define HEPS 1e-7f

__global__ void sd_init_max(int* maxbits) {
    if (threadIdx.x == 0) *maxbits = 0;  // bits of 0.0f; all inv_dist > 0
}

// One wave handles 16 edges via V_WMMA_F32_16X16X4_F32 over 16 K-steps.
// D = A(16x64 src rows) x B(64x16 dst cols); diagonal = per-edge Lorentzian dots.
__global__ __launch_bounds__(256) void sd_edge_invdist(
    const float* __restrict__ z, const int* __restrict__ edge_index,
    float* __restrict__ inv_out, int* __restrict__ maxbits, int n_edges)
{
    const int lane = threadIdx.x & 31;
    const int wave = threadIdx.x >> 5;
    const long long base = ((long long)blockIdx.x * 8 + wave) * 16;  // 16 edges/wave
    if (base >= n_edges) return;  // wave-uniform: scalar branch, EXEC stays all-ones

    const int row  = lane & 15;   // A row / B column index within the 16-edge group
    const int half = lane >> 4;   // lane half selects K pair {2h, 2h+1}

    long long e  = base + row;
    long long ec = (e < n_edges) ? e : (long long)(n_edges - 1);  // clamp tail loads
    const int srcRow = edge_index[ec];
    const int dstRow = edge_index[(long long)n_edges + ec];

    const float* __restrict__ aptr = z + (long long)srcRow * DIM + 2 * half;
    const float* __restrict__ bptr = z + (long long)dstRow * DIM + 2 * half;

    v8f acc = {};
#pragma unroll
    for (int k0 = 0; k0 < DIM; k0 += 4) {
        v2f a = *(const v2f*)(aptr + k0);   // A: VGPR0=K(2h), VGPR1=K(2h+1)
        v2f b = *(const v2f*)(bptr + k0);   // B: symmetric layout
        if (k0 == 0 && half == 0) a.x = -a.x;  // Lorentz sign on coordinate 0
        acc = __builtin_amdgcn_wmma_f32_16x16x4_f32(
            /*neg_a=*/false, a, /*neg_b=*/false, b,
            /*c_mod=*/(short)0, acc, /*reuse_a=*/false, /*reuse_b=*/false);
    }

    // Diagonal (m,m): m<8 -> lane m, acc[m]; m>=8 -> lane m+16, acc[m-8].
    const int j = lane & 7;
    float d = acc[0];
    d = (j == 1) ? acc[1] : d;
    d = (j == 2) ? acc[2] : d;
    d = (j == 3) ? acc[3] : d;
    d = (j == 4) ? acc[4] : d;
    d = (j == 5) ? acc[5] : d;
    d = (j == 6) ? acc[6] : d;
    d = (j == 7) ? acc[7] : d;
    const int srcLane = ((lane & 8) ? (lane + 16) : lane) & 31;
    const float dot = __shfl(d, srcLane, 32);  // lanes 0-15 now hold edge dots

    float red = -INFINITY;
    if (lane < 16 && (base + lane) < n_edges) {
        // inner = -x0*y0 + sum(xi*yi) == dot (sign folded in). radius = 1.
        float arg  = fmaxf(-dot, 1.0f + HEPS);
        float dist = logf(arg + sqrtf(arg * arg - 1.0f));  // arccosh
        float inv  = 1.0f / dist;
        inv_out[base + lane] = inv;
        red = inv;
    }

    // Wave max (positive floats: int-bit compare is order-preserving).
#pragma unroll
    for (int off = 16; off >= 1; off >>= 1)
        red = fmaxf(red, __shfl_xor(red, off, 32));
    if (lane == 0)
        atomicMax(maxbits, __float_as_int(red));
}

// probs = sigmoid(p)*softmax(inv) / max(sigmoid(p)*softmax(inv))
//       = exp(inv - max(inv))   (scalar and denominator cancel exactly)
__global__ __launch_bounds__(256) void sd_finalize(
    const float* __restrict__ inv, const int* __restrict__ maxbits,
    float* __restrict__ out, int n_edges)
{
    int i = blockIdx.x * blockDim.x + threadIdx.x;
    if (i < n_edges) {
        float m = __int_as_float(*maxbits);
        out[i] = expf(inv[i] - m);
    }
}

extern "C" void kernel_launch(void* const* d_in, const int* in_sizes, int n_in,
                              void* d_out, int out_size, void* d_ws, size_t ws_size,
                              hipStream_t stream) {
    const float* z          = (const float*)d_in[0];
    // d_in[1] is p: sigmoid(p) cancels in probs = scores/max(scores).
    const int*   edge_index = (const int*)d_in[2];
    const int    n_edges    = in_sizes[2] / 2;
    float*       out        = (float*)d_out;

    float* inv_ws  = (float*)d_ws;
    size_t maxOff  = (((size_t)n_edges * sizeof(float)) + 255) & ~(size_t)255;
    int*   maxbits = (int*)((char*)d_ws + maxOff);

    sd_init_max<<<1, 32, 0, stream>>>(maxbits);

    int groups = (n_edges + 15) / 16;   // 16 edges per wave
    int blocks = (groups + 7) / 8;      // 8 waves (256 threads) per block
    sd_edge_invdist<<<blocks, 256, 0, stream>>>(z, edge_index, inv_ws, maxbits, n_edges);

    sd_finalize<<<(n_edges + 255) / 256, 256, 0, stream>>>(inv_ws, maxbits, out, n_edges);
}